// FWQ_70360154243709
// MI455X (gfx1250) — compile-verified
//
#include <hip/hip_runtime.h>
#include <math.h>

typedef __attribute__((ext_vector_type(2))) float v2f;
typedef __attribute__((ext_vector_type(8))) float v8f;

#define LN2F        0.69314718055994530942f
#define CBRT_2_3    0.87358046473629588f   /* (2/3)^(1/3) */
#define INV_CBRT18  0.38157141418444388f   /* 1/18^(1/3) = 1/(2^(1/3)*3^(2/3)) */
#define CAVAF       62.0f
#define TOLF        0.0001f
#define QEPF        200.0f

// Q(u) = cbrt(2/3)*u/v + v/cbrt(18) + 1, clipped to [2, 2^32].
// Clip is written comparison-style so NaN propagates (matches jnp.clip).
__device__ __forceinline__ float Qfun(float u) {
    float v = cbrtf(u * sqrtf(81.0f - 12.0f * u) + 9.0f * u);
    float Q = CBRT_2_3 * (u / v) + v * INV_CBRT18 + 1.0f;
    Q = (Q < 2.0f) ? 2.0f : Q;
    Q = (Q > 4294967296.0f) ? 4294967296.0f : Q;
    return Q;
}

// _uq_idx(x,lo,hi,Q)/div*(hi-lo)+lo   with jnp.round == rintf (half-to-even)
__device__ __forceinline__ float uq_deq(float x, float lo, float hi, float Q, float div) {
    float xc  = fminf(fmaxf(x, lo), hi);
    float idx = rintf((xc - lo) / (hi - lo) * (Q - 1.0f));
    return idx / div * (hi - lo) + lo;
}

// ---------------- Pass 1: per-chunk column sum (via WMMA ones-GEMM) + min/max
__global__ void __launch_bounds__(256)
afq_colstat_wmma(const float* __restrict__ x,
                 float* __restrict__ sumPart,
                 float* __restrict__ minPart,
                 float* __restrict__ maxPart,
                 int Nn, int Dn, int rowsPerChunk)
{
    const int tid   = threadIdx.x;
    const int wave  = tid >> 5;
    const int lane  = tid & 31;
    const int n     = lane & 15;       // column within 16-wide group (B: N = lane%16)
    const int khalf = lane >> 4;       // which K-half this lane supplies
    const int col   = blockIdx.x * 128 + wave * 16 + n;
    const int chunk = blockIdx.y;

    const int r0 = chunk * rowsPerChunk;
    int r1 = r0 + rowsPerChunk; if (r1 > Nn) r1 = Nn;

    v2f a; a.x = 1.0f; a.y = 1.0f;     // ones matrix: layout-invariant
    v8f c = {};
    float mn = INFINITY, mx = -INFINITY;

    for (int r = r0; r + 4 <= r1; r += 4) {
        const float* p = x + (long long)(r + 2 * khalf) * Dn + col;
        v2f b; b.x = p[0]; b.y = p[Dn];          // rows (r+2kh), (r+2kh+1), col n
        mn = fminf(mn, fminf(b.x, b.y));
        mx = fmaxf(mx, fmaxf(b.x, b.y));
        // D = ones(16x4) * B(4x16) + C  -> every D row holds the column sums
        c = __builtin_amdgcn_wmma_f32_16x16x4_f32(false, a, false, b,
                                                  (short)0, c, false, false);
    }
    // the two K-half lane groups covered disjoint rows of the same column
    mn = fminf(mn, __shfl_xor(mn, 16, 32));
    mx = fmaxf(mx, __shfl_xor(mx, 16, 32));

    if (lane < 16) {
        long long o = (long long)chunk * Dn + col;
        sumPart[o] = c[0];             // D[0][n] = column sum for this chunk
        minPart[o] = mn;
        maxPart[o] = mx;
    }
}

// ---------------- Pass 2: reduce chunk partials -> per-column min/max/mean/range
__global__ void afq_colstat_reduce(const float* __restrict__ sumPart,
                                   const float* __restrict__ minPart,
                                   const float* __restrict__ maxPart,
                                   float* __restrict__ colMin,
                                   float* __restrict__ colMax,
                                   float* __restrict__ colMean,
                                   float* __restrict__ colRange,
                                   int Dn, int chunks, int Nn)
{
    int j = blockIdx.x * blockDim.x + threadIdx.x;
    if (j >= Dn) return;
    float s = 0.0f, mn = INFINITY, mx = -INFINITY;
    for (int cth = 0; cth < chunks; ++cth) {
        long long o = (long long)cth * Dn + j;
        s += sumPart[o];
        mn = fminf(mn, minPart[o]);
        mx = fmaxf(mx, maxPart[o]);
    }
    colMin[j] = mn; colMax[j] = mx;
    colMean[j] = s / (float)Nn;
    colRange[j] = mx - mn;
}

// ---------------- block reductions (1024 threads, fixed tree: deterministic)
__device__ __forceinline__ float blkSum(float v, float* red, int tid) {
    red[tid] = v; __syncthreads();
    for (int s = 512; s > 0; s >>= 1) { if (tid < s) red[tid] += red[tid + s]; __syncthreads(); }
    float r = red[0]; __syncthreads(); return r;
}
__device__ __forceinline__ float blkMin(float v, float* red, int tid) {
    red[tid] = v; __syncthreads();
    for (int s = 512; s > 0; s >>= 1) { if (tid < s) red[tid] = fminf(red[tid], red[tid + s]); __syncthreads(); }
    float r = red[0]; __syncthreads(); return r;
}
__device__ __forceinline__ float blkMax(float v, float* red, int tid) {
    red[tid] = v; __syncthreads();
    for (int s = 512; s > 0; s >>= 1) { if (tid < s) red[tid] = fmaxf(red[tid], red[tid + s]); __syncthreads(); }
    float r = red[0]; __syncthreads(); return r;
}

// ---------------- Pass 3: ranks + 8-candidate bisection + per-column params
__global__ void __launch_bounds__(1024)
afq_solve(const float* __restrict__ colMin, const float* __restrict__ colMax,
          const float* __restrict__ colMean, const float* __restrict__ colRange,
          float* __restrict__ pTwo, float* __restrict__ pMinQ,
          float* __restrict__ pMaxQ, float* __restrict__ pQe,
          float* __restrict__ pMeanQ, int Dn, int Nn)
{
    __shared__ float s_rng[1024];
    __shared__ float s_red[1024];
    __shared__ float s_lo, s_hi, s_mid;
    __shared__ int   s_it, s_done;
    __shared__ float s_bestErr, s_bestNu;
    __shared__ int   s_bestM;

    const int tid = threadIdx.x;          // one thread per column, Dn == 1024
    const float myRange = colRange[tid];
    const float myMean  = colMean[tid];
    const float myMin   = colMin[tid];
    const float myMax   = colMax[tid];
    s_rng[tid] = myRange;
    __syncthreads();

    // stable rank under argsort(-ranges)
    int rank = 0;
    for (int k = 0; k < Dn; ++k) {
        float rk = s_rng[k];
        rank += (rk > myRange) || (rk == myRange && k < tid);
    }

    float meanMax = blkMax(myMean, s_red, tid);
    float meanMin = blkMin(myMean, s_red, tid);
    const float a0 = meanMax - meanMin;
    const float Bf = (float)Nn;

    if (tid == 0) { s_bestErr = INFINITY; s_bestM = 1; s_bestNu = 1.0f; }
    __syncthreads();

    int prevM = -1;
    for (int k = 0; k < 8; ++k) {
        int M = (int)(1.0 + (double)k * (double)((Dn / 2) - 1) / 7.0); // np.linspace(...,int)
        if (M == prevM) continue;                                      // np.unique
        prevM = M;

        if (tid == 0) { s_lo = 1e-12f; s_hi = 1000000.0f; s_mid = 0.0f; s_it = 0; s_done = 0; }
        __syncthreads();
        for (;;) {
            if (s_it >= 100 || s_done) break;       // uniform: all read same shared state
            float mid = 0.5f * (s_lo + s_hi);
            float lg = 0.0f;
            if (rank < M) lg = log2f(Qfun(myRange * myRange * LN2F / (2.0f * mid)));
            float bs = blkSum(lg, s_red, tid);
            if (tid == 0) {
                bs += log2f(Qfun(a0 * a0 * Bf * LN2F / mid));
                if (bs > CAVAF) s_lo = mid; else s_hi = mid;   // NaN -> hi=mid (matches jnp)
                s_mid = mid; s_it = s_it + 1;
                s_done = (fabsf(bs - CAVAF) < TOLF) ? 1 : 0;
            }
            __syncthreads();
        }
        const float nu = s_mid;
        float Qj = Qfun(myRange * myRange * LN2F / (2.0f * nu));
        float contrib = (rank < M)
            ? (myRange * myRange * Bf / (4.0f * (Qj - 1.0f) * (Qj - 1.0f)))
            : (myRange * myRange * Bf * 0.5f);
        float err = blkSum(contrib, s_red, tid);
        if (tid == 0) {
            float Q0 = Qfun(a0 * a0 * Bf * LN2F / nu);
            err += a0 * a0 * Bf * (float)(Dn - M) / (2.0f * (Q0 - 1.0f) * (Q0 - 1.0f));
            if (err < s_bestErr) { s_bestErr = err; s_bestM = M; s_bestNu = nu; } // first argmin
        }
        __syncthreads();
    }

    const int   M_star = s_bestM;
    const float nuS    = s_bestNu;
    const int   two    = (rank < M_star) ? 1 : 0;

    float QjS = Qfun(myRange * myRange * LN2F / (2.0f * nuS));
    float Q0S = Qfun(a0 * a0 * Bf * LN2F / nuS);
    float Qe  = rintf(QjS);
    float Q0r = rintf(Q0S);

    float min_low = blkMin(two ? myMin :  INFINITY, s_red, tid);
    float min_up  = blkMax(two ? myMin : -INFINITY, s_red, tid);
    float max_low = blkMin(two ? myMax :  INFINITY, s_red, tid);
    float max_up  = blkMax(two ? myMax : -INFINITY, s_red, tid);
    float qmin    = blkMin(two ?  INFINITY : myMean, s_red, tid);
    float qmax    = blkMax(two ? -INFINITY : myMean, s_red, tid);

    float minq  = uq_deq(myMin,  min_low, min_up, QEPF, QEPF - 1.0f);
    float maxq  = uq_deq(myMax,  max_low, max_up, QEPF, QEPF - 1.0f);
    float meanq = uq_deq(myMean, qmin,    qmax,   Q0r,  Q0r);

    pTwo[tid]  = (float)two;
    pMinQ[tid] = minq;
    pMaxQ[tid] = maxq;
    pQe[tid]   = Qe;
    pMeanQ[tid]= meanq;
}

// ---------------- Pass 4: streaming requantize (float4), + trailing zeros
__global__ void __launch_bounds__(256)
afq_apply(const float* __restrict__ x,
          const float* __restrict__ pTwo, const float* __restrict__ pMinQ,
          const float* __restrict__ pMaxQ, const float* __restrict__ pQe,
          const float* __restrict__ pMeanQ,
          float* __restrict__ out, long long total, int Dn)
{
    long long g = (long long)blockIdx.x * blockDim.x + threadIdx.x;
    long long e = g * 4;
    if (e >= total) return;
    if (e + 16384 < total) __builtin_prefetch(x + e + 16384, 0, 0); // global_prefetch_b8

    const int j = (int)(e % Dn);      // e and Dn both multiples of 4 -> j+3 < Dn
    const float4 xv = *reinterpret_cast<const float4*>(x + e);
    const float xs[4] = { xv.x, xv.y, xv.z, xv.w };
    float r[4];
#pragma unroll
    for (int cidx = 0; cidx < 4; ++cidx) {
        int jc = j + cidx;
        float lo = pMinQ[jc], hi = pMaxQ[jc], Qe = pQe[jc];
        float xc = fminf(fmaxf(xs[cidx], lo), hi);
        float q  = rintf((xc - lo) / (hi - lo) * (Qe - 1.0f)) / Qe * (hi - lo) + lo;
        r[cidx]  = (pTwo[jc] != 0.0f) ? q : pMeanQ[jc];
    }
    float4 ov; ov.x = r[0]; ov.y = r[1]; ov.z = r[2]; ov.w = r[3];
    *reinterpret_cast<float4*>(out + e) = ov;

    if (g == 0) { out[total] = 0.0f; out[total + 1] = 0.0f; }  // the two scalar 0.0 outputs
}

extern "C" void kernel_launch(void* const* d_in, const int* in_sizes, int n_in,
                              void* d_out, int out_size, void* d_ws, size_t ws_size,
                              hipStream_t stream)
{
    (void)n_in; (void)out_size; (void)ws_size;
    const float* x  = (const float*)d_in[0];
    float* out      = (float*)d_out;
    const long long total = (long long)in_sizes[0];   // 8*2048*1024
    const int Dn = 1024;
    const int Nn = (int)(total / Dn);                 // 16384
    const int CHUNKS = 32;
    int rowsPerChunk = ((Nn + CHUNKS - 1) / CHUNKS + 3) & ~3;

    float* ws       = (float*)d_ws;
    float* sumPart  = ws;                          // CHUNKS*Dn
    float* minPart  = sumPart + (size_t)CHUNKS * Dn;
    float* maxPart  = minPart + (size_t)CHUNKS * Dn;
    float* colMin   = maxPart + (size_t)CHUNKS * Dn;
    float* colMax   = colMin  + Dn;
    float* colMean  = colMax  + Dn;
    float* colRange = colMean + Dn;
    float* pTwo     = colRange+ Dn;
    float* pMinQ    = pTwo    + Dn;
    float* pMaxQ    = pMinQ   + Dn;
    float* pQe      = pMaxQ   + Dn;
    float* pMeanQ   = pQe     + Dn;

    dim3 g1(Dn / 128, CHUNKS);                 // 8 waves/block * 16 cols = 128 cols/block
    afq_colstat_wmma<<<g1, 256, 0, stream>>>(x, sumPart, minPart, maxPart,
                                             Nn, Dn, rowsPerChunk);

    afq_colstat_reduce<<<(Dn + 255) / 256, 256, 0, stream>>>(
        sumPart, minPart, maxPart, colMin, colMax, colMean, colRange,
        Dn, CHUNKS, Nn);

    afq_solve<<<1, 1024, 0, stream>>>(colMin, colMax, colMean, colRange,
                                      pTwo, pMinQ, pMaxQ, pQe, pMeanQ, Dn, Nn);

    long long groups = total / 4;
    afq_apply<<<(unsigned)((groups + 255) / 256), 256, 0, stream>>>(
        x, pTwo, pMinQ, pMaxQ, pQe, pMeanQ, out, total, Dn);
}